// Beam_57612691308621
// MI455X (gfx1250) — compile-verified
//
#include <hip/hip_runtime.h>
#include <hip/hip_bf16.h>
#include <stdint.h>

// Problem constants (from reference setup_inputs)
#define P_ 32
#define D_ 8
#define S_ 2048
#define V_ 128000
#define CPB 5                      // chunks per beam
#define CH_ (V_ / CPB)             // 25600 floats per chunk
#define CHUNKS_ (D_ * CPB)         // 40 chunks per prompt
#define TPB1 256
#define TILES_ (CH_ / (TPB1 * 4))  // 25 tiles of 256 float4
#define KK 16
#define EOS_ 2
#define INF_ 10000000.0f

// d_out layout (floats, concatenated in return order)
#define OFF_ATT  ((size_t)0)
#define OFF_ASEQ ((size_t)P_ * D_)
#define OFF_ALP  (OFF_ASEQ + (size_t)P_ * D_ * S_)
#define OFF_FSEQ (OFF_ALP + (size_t)P_ * D_)
#define OFF_FLP  (OFF_FSEQ + (size_t)P_ * D_ * S_)

// ---------- CDNA5 async global->LDS path (guarded; falls back to b128 loads) ----
#ifdef __has_builtin
#if __has_builtin(__builtin_amdgcn_global_load_async_to_lds_b128)
#define HAVE_ASYNC_LDS 1
#endif
#endif
#ifndef HAVE_ASYNC_LDS
#define HAVE_ASYNC_LDS 0
#endif

#if HAVE_ASYNC_LDS
typedef int v4i __attribute__((ext_vector_type(4)));
typedef __attribute__((address_space(1))) v4i GV4;  // global (target AS1)
typedef __attribute__((address_space(3))) v4i LV4;  // LDS (target AS3)

__device__ __forceinline__ void async_ld_b128(const void* g, void* l) {
  // Cross-AS pointer casts must go through an integer in clang.
  // generic->AS1: flat global address == global address.
  // generic->AS3: flat LDS address low 32 bits == LDS byte offset (ISA 10.2).
  GV4* gp = (GV4*)(uintptr_t)g;
  LV4* lp = (LV4*)(uintptr_t)l;
  __builtin_amdgcn_global_load_async_to_lds_b128(gp, lp, 0, 0);
}
__device__ __forceinline__ void wait_async_le1() {
#if __has_builtin(__builtin_amdgcn_s_wait_asynccnt)
  __builtin_amdgcn_s_wait_asynccnt(1);
#else
  asm volatile("s_wait_asynccnt 0x1" ::: "memory");
#endif
}
__device__ __forceinline__ void wait_async_le0() {
#if __has_builtin(__builtin_amdgcn_s_wait_asynccnt)
  __builtin_amdgcn_s_wait_asynccnt(0);
#else
  asm volatile("s_wait_asynccnt 0x0" ::: "memory");
#endif
}
#endif

// Insert key into ascending top-16 kept in registers (static indices only).
__device__ __forceinline__ void topk_insert(unsigned long long (&A)[KK],
                                            unsigned long long k) {
  if (k > A[0]) {
    A[0] = k;
#pragma unroll
    for (int j = 0; j < KK - 1; ++j) {
      unsigned long long a = A[j], b = A[j + 1];
      A[j]     = a < b ? a : b;
      A[j + 1] = a < b ? b : a;
    }
  }
}

// ============================================================================
// Kernel 1: streaming scan. One block per (prompt, chunk). Chunk = one beam
// slice of 25600 probs. Per-thread reg top-16 on raw prob (monotone w/ log),
// wave tournament merge, lane0 final merge -> chunk top-16 u64 keys to ws.
// key = (prob_bits << 32) | ~flat_idx   (flat_idx = beam*V + v), so ties pick
// the lower index like jax.lax.top_k.
// ============================================================================
__global__ __launch_bounds__(TPB1) void k_scan(
    const float* __restrict__ probs, unsigned long long* __restrict__ part) {
#if HAVE_ASYNC_LDS
  __shared__ float4 buf[2][TPB1];
#endif
  __shared__ unsigned long long wl[8 * KK];

  const int tid  = threadIdx.x;
  const int lane = tid & 31;
  const int wv   = tid >> 5;
  const int b    = blockIdx.x;
  const int p    = b / CHUNKS_;
  const int c    = b % CHUNKS_;
  const int beam = c / CPB;
  const int sub  = c % CPB;

  const size_t gbase   = ((size_t)(p * D_ + beam)) * V_ + (size_t)sub * CH_;
  const unsigned fbase = (unsigned)(beam * V_ + sub * CH_);
  const float4* gp     = (const float4*)(probs + gbase);

  unsigned long long arr[KK];
#pragma unroll
  for (int i = 0; i < KK; ++i) arr[i] = 0ull;
  float thr = 0.0f;  // prob of current 16th-best (probs are > 0)

  auto ins = [&](float v, unsigned idx) {
    if (v > thr) {
      unsigned long long k =
          ((unsigned long long)__float_as_uint(v) << 32) | (unsigned)(~idx);
      topk_insert(arr, k);
      thr = __uint_as_float((unsigned)(arr[0] >> 32));
    }
  };

#if HAVE_ASYNC_LDS
  async_ld_b128(gp + tid, &buf[0][tid]);  // prime tile 0
#endif
#pragma unroll 1
  for (int t = 0; t < TILES_; ++t) {
    float4 v;
#if HAVE_ASYNC_LDS
    if (t + 1 < TILES_)
      async_ld_b128(gp + (t + 1) * TPB1 + tid, &buf[(t + 1) & 1][tid]);
    if (t + 1 < TILES_) wait_async_le1(); else wait_async_le0();
    v = buf[t & 1][tid];  // each thread reads back only its own slot
#else
    v = gp[t * TPB1 + tid];
#endif
    const unsigned ib = fbase + (unsigned)(t * TPB1 + tid) * 4u;
    ins(v.x, ib);
    ins(v.y, ib + 1u);
    ins(v.z, ib + 2u);
    ins(v.w, ib + 3u);
  }

  // Wave tournament: 16 rounds of pop-max across 32 lanes (static indices).
#pragma unroll 1
  for (int r = 0; r < KK; ++r) {
    unsigned long long cand = arr[KK - 1];
    unsigned long long m = cand;
#pragma unroll
    for (int off = 16; off >= 1; off >>= 1) {
      unsigned long long o = __shfl_xor(m, off, 32);
      m = o > m ? o : m;
    }
    unsigned long long bal = __ballot(cand == m);
    int w = (int)__ffsll(bal) - 1;
    if (lane == w) {  // pop my max
#pragma unroll
      for (int j = KK - 1; j > 0; --j) arr[j] = arr[j - 1];
      arr[0] = 0ull;
    }
    if (lane == 0) wl[wv * KK + r] = m;
  }
  __syncthreads();

  if (tid == 0) {
    unsigned long long fin[KK];
#pragma unroll
    for (int i = 0; i < KK; ++i) fin[i] = 0ull;
    for (int i = 0; i < 8 * KK; ++i) topk_insert(fin, wl[i]);
    unsigned long long* o = part + ((size_t)p * CHUNKS_ + c) * KK;
#pragma unroll
    for (int i = 0; i < KK; ++i) o[i] = fin[KK - 1 - i];  // descending
  }
}

// Order-preserving map for float -> u32 (handles negatives / -inf).
__device__ __forceinline__ unsigned fkey(float f) {
  unsigned b = __float_as_uint(f);
  return (b & 0x80000000u) ? ~b : (b | 0x80000000u);
}
__device__ __forceinline__ float funkey(unsigned m) {
  unsigned b = (m & 0x80000000u) ? (m & 0x7FFFFFFFu) : ~m;
  return __uint_as_float(b);
}

// ============================================================================
// Kernel 2: per-prompt merge + all selection logic. One block per prompt.
// 640 candidates get logf + alive_lp offset (parallel), then lane 0 runs the
// tiny serial top-16 / top-8 selection (matches jax top_k stability).
// Writes: attention ids, new_alive_lp, new_fin_lp to d_out; gather tables to ws.
// ============================================================================
__global__ __launch_bounds__(256) void k_select(
    const unsigned long long* __restrict__ part,
    const float* __restrict__ alive_lp_in, const float* __restrict__ fin_lp_in,
    const int* __restrict__ still_prompt, const int* __restrict__ is_first,
    float* __restrict__ out, int* __restrict__ tbl) {
  __shared__ unsigned long long e[CHUNKS_ * KK];  // 640 keys, lp-ordered
  const int p = blockIdx.x, tid = threadIdx.x;

  for (int i = tid; i < CHUNKS_ * KK; i += 256) {
    unsigned long long k = part[(size_t)p * CHUNKS_ * KK + i];
    float prob   = __uint_as_float((unsigned)(k >> 32));
    unsigned idx = ~(unsigned)k;                 // flat idx in [0, D*V)
    int beam     = (int)(idx / (unsigned)V_);
    float lp     = alive_lp_in[p * D_ + beam] + logf(prob);
    e[i] = ((unsigned long long)fkey(lp) << 32) | (unsigned)(~idx);
  }
  __syncthreads();
  if (tid != 0) return;

  // top16 overall (640) and top16 of beam-0 entries (first CPB*KK = 80)
  unsigned long long ov[KK], d0[KK];
#pragma unroll
  for (int i = 0; i < KK; ++i) { ov[i] = 0ull; d0[i] = 0ull; }
  for (int i = 0; i < CHUNKS_ * KK; ++i) topk_insert(ov, e[i]);
  for (int i = 0; i < CPB * KK; ++i)     topk_insert(d0, e[i]);

  const bool sp = still_prompt[p] != 0;
  const bool fi = is_first[p] != 0;

  float lp16[KK]; int beam16[KK], tok16[KK];
#pragma unroll
  for (int j = 0; j < KK; ++j) {
    unsigned long long k = ov[KK - 1 - j];        // descending
    unsigned idx = ~(unsigned)k;
    beam16[j]    = (int)(idx / (unsigned)V_);     // NOT overridden by is_first
    if (fi) {
      unsigned long long k0 = d0[KK - 1 - j];
      unsigned idx0 = ~(unsigned)k0;              // beam 0 => idx0 == token
      lp16[j]  = funkey((unsigned)(k0 >> 32));    // alive_lp[p,0] + log(prob)
      tok16[j] = (int)(idx0 % (unsigned)V_);
    } else {
      lp16[j]  = funkey((unsigned)(k >> 32));
      tok16[j] = (int)(idx % (unsigned)V_);
    }
  }

  float am[KK], fm[KK];
#pragma unroll
  for (int j = 0; j < KK; ++j) {
    const bool fin = (tok16[j] == EOS_);
    am[j] = lp16[j] + (fin ? -INF_ : 0.0f);
    fm[j] = lp16[j] + (fin ? 0.0f : -INF_);
  }

  // ---- new alive: stable top-8 of am ----
  bool used[KK];
#pragma unroll
  for (int j = 0; j < KK; ++j) used[j] = false;
  for (int i = 0; i < D_; ++i) {
    int bj = 0; float bv = -3.4e38f;
    for (int j = 0; j < KK; ++j)
      if (!used[j] && am[j] > bv) { bv = am[j]; bj = j; }
    used[bj] = true;
    out[OFF_ATT + (size_t)p * D_ + i] = sp ? (float)i : (float)beam16[bj];
    out[OFF_ALP + (size_t)p * D_ + i] = sp ? alive_lp_in[p * D_ + i] : bv;
    tbl[p * 40 + i]      = sp ? i : beam16[bj];   // source beam row
    tbl[p * 40 + 8 + i]  = sp ? -1 : tok16[bj];   // token to splice (-1 = none)
  }

  // ---- new finished: stable top-8 over [fin_lp(8) | fm(16)] ----
  float cv[D_ + KK];
  for (int i = 0; i < D_; ++i) cv[i] = fin_lp_in[p * D_ + i];
  for (int j = 0; j < KK; ++j) cv[D_ + j] = fm[j];
  bool used2[D_ + KK];
  for (int j = 0; j < D_ + KK; ++j) used2[j] = false;
  for (int i = 0; i < D_; ++i) {
    int bj = 0; float bv = -3.4e38f;
    for (int j = 0; j < D_ + KK; ++j)
      if (!used2[j] && cv[j] > bv) { bv = cv[j]; bj = j; }
    used2[bj] = true;
    out[OFF_FLP + (size_t)p * D_ + i] = sp ? fin_lp_in[p * D_ + i] : bv;
    int isOld, row, tok;
    if (sp)            { isOld = 1; row = i;               tok = -1; }
    else if (bj < D_)  { isOld = 1; row = bj;              tok = -1; }
    else               { isOld = 0; row = beam16[bj - D_]; tok = tok16[bj - D_]; }
    tbl[p * 40 + 16 + i] = isOld;
    tbl[p * 40 + 24 + i] = row;
    tbl[p * 40 + 32 + i] = tok;
  }
}

// ============================================================================
// Kernel 3: gather sequence rows. grid = (16 rows, P prompts).
// Rows 0..7 -> new_alive_seq, 8..15 -> new_fin_seq. b128 loads, splice
// cur_pos token, convert int -> float for the float d_out.
// ============================================================================
__global__ __launch_bounds__(256) void k_copy(
    const int* __restrict__ alive_seq, const int* __restrict__ fin_seq,
    const int* __restrict__ tbl, const int* __restrict__ cur_pos,
    float* __restrict__ out) {
  const int p = blockIdx.y, j = blockIdx.x, tid = threadIdx.x;
  const int* t = tbl + p * 40;
  const int cp = cur_pos[0];

  const int* src; float* dst; int tok;
  if (j < D_) {
    int bm = t[j]; tok = t[8 + j];
    src = alive_seq + ((size_t)(p * D_ + bm)) * S_;
    dst = out + OFF_ASEQ + ((size_t)(p * D_ + j)) * S_;
  } else {
    int jj = j - D_;
    int isOld = t[16 + jj], row = t[24 + jj]; tok = t[32 + jj];
    src = (isOld ? fin_seq : alive_seq) + ((size_t)(p * D_ + row)) * S_;
    dst = out + OFF_FSEQ + ((size_t)(p * D_ + jj)) * S_;
  }

  for (int e0 = tid * 4; e0 < S_; e0 += 256 * 4) {
    int4 v = *(const int4*)(src + e0);
    if (tok >= 0) {
      if (cp == e0)     v.x = tok;
      if (cp == e0 + 1) v.y = tok;
      if (cp == e0 + 2) v.z = tok;
      if (cp == e0 + 3) v.w = tok;
    }
    float4 f = make_float4((float)v.x, (float)v.y, (float)v.z, (float)v.w);
    *(float4*)(dst + e0) = f;
  }
}

extern "C" void kernel_launch(void* const* d_in, const int* in_sizes, int n_in,
                              void* d_out, int out_size, void* d_ws,
                              size_t ws_size, hipStream_t stream) {
  (void)in_sizes; (void)n_in; (void)out_size; (void)ws_size;
  const float* probs        = (const float*)d_in[0];
  const int*   alive_seq    = (const int*)d_in[1];
  const int*   fin_seq      = (const int*)d_in[2];
  const float* alive_lp     = (const float*)d_in[3];
  const float* fin_lp       = (const float*)d_in[4];
  const int*   still_prompt = (const int*)d_in[5];
  const int*   is_first     = (const int*)d_in[6];
  const int*   cur_pos      = (const int*)d_in[7];

  unsigned long long* part = (unsigned long long*)d_ws;
  int* tbl = (int*)((char*)d_ws +
                    sizeof(unsigned long long) * (size_t)P_ * CHUNKS_ * KK);
  float* out = (float*)d_out;

  k_scan<<<P_ * CHUNKS_, TPB1, 0, stream>>>(probs, part);
  k_select<<<P_, 256, 0, stream>>>(part, alive_lp, fin_lp, still_prompt,
                                   is_first, out, tbl);
  dim3 g3(16, P_);
  k_copy<<<g3, 256, 0, stream>>>(alive_seq, fin_seq, tbl, cur_pos, out);
}